// FasterRCNN_30236569764211
// MI455X (gfx1250) — compile-verified
//
#include <hip/hip_runtime.h>
#include <math.h>

#define N_IMG   8
#define A_ANCH  1000000
#define PRE_K   2000
#define POST_K  1000
#define K_PAD   2048
#define CAND    8192
#define BINS    4096
#define SUPW    64
#define IOU_THR 0.7f
#define MIN_SZ  1e-3f
#define IMG_SZ  800.0f
#define BBOX_CLIP 4.135166556742356f  /* log(1000/16) */

/* ---- gfx1250 async global->LDS (guarded; falls back to ds stores) ---- */
#if defined(__has_builtin)
# if __has_builtin(__builtin_amdgcn_global_load_async_to_lds_b64) && \
     __has_builtin(__builtin_amdgcn_s_wait_asynccnt)
#  define HAVE_ASYNC_LDS 1
# endif
# if __has_builtin(__builtin_amdgcn_wmma_f32_16x16x4_f32)
#  define HAVE_WMMA4 1
# endif
#endif
#ifndef HAVE_ASYNC_LDS
# define HAVE_ASYNC_LDS 0
#endif
#ifndef HAVE_WMMA4
# define HAVE_WMMA4 0
#endif

typedef int   async_v2i __attribute__((vector_size(8)));
typedef float v2f __attribute__((ext_vector_type(2)));
typedef float v8f __attribute__((ext_vector_type(8)));
#define AS1 __attribute__((address_space(1)))
#define AS3 __attribute__((address_space(3)))

/* order-preserving float<->uint key (ascending) */
__device__ __forceinline__ unsigned key_of(float f) {
  unsigned u = __float_as_uint(f);
  return u ^ (((unsigned)(((int)u) >> 31)) | 0x80000000u);
}
__device__ __forceinline__ float key_to_float(unsigned k) {
  unsigned u = (k & 0x80000000u) ? (k ^ 0x80000000u) : ~k;
  return __uint_as_float(u);
}

/* ---------------- K0: zero histogram + counters ---------------- */
__global__ void k0_init(unsigned* __restrict__ hist, unsigned* __restrict__ cnt) {
  int i = blockIdx.x * blockDim.x + threadIdx.x;
  if (i < N_IMG * BINS) hist[i] = 0u;
  if (i < N_IMG) cnt[i] = 0u;
}

/* ---------------- K1: per-image 4096-bin histogram ---------------- */
__global__ __launch_bounds__(256) void k1_hist(const float* __restrict__ obj,
                                               unsigned* __restrict__ hist) {
  __shared__ unsigned sh[BINS];
  const int img = blockIdx.y;
  for (int b = threadIdx.x; b < BINS; b += blockDim.x) sh[b] = 0u;
  __syncthreads();
  const float4* src = (const float4*)(obj + (size_t)img * A_ANCH);
  const int n4 = A_ANCH / 4;
  const int stride = gridDim.x * blockDim.x;
  for (int i = blockIdx.x * blockDim.x + threadIdx.x; i < n4; i += stride) {
    __builtin_prefetch(src + i + 2 * stride, 0, 0);   /* global_prefetch_b8 */
    float4 v = src[i];
    atomicAdd(&sh[key_of(v.x) >> 20], 1u);
    atomicAdd(&sh[key_of(v.y) >> 20], 1u);
    atomicAdd(&sh[key_of(v.z) >> 20], 1u);
    atomicAdd(&sh[key_of(v.w) >> 20], 1u);
  }
  __syncthreads();
  unsigned* gh = hist + (size_t)img * BINS;
  for (int b = threadIdx.x; b < BINS; b += blockDim.x) {
    unsigned c = sh[b];
    if (c) atomicAdd(&gh[b], c);
  }
}

/* ---------------- K2: find threshold bin per image ---------------- */
__global__ __launch_bounds__(512) void k2_thresh(const unsigned* __restrict__ hist,
                                                 unsigned* __restrict__ thr) {
  __shared__ unsigned part[512];
  __shared__ int best;
  const int img = blockIdx.x, t = threadIdx.x;
  const unsigned* h = hist + (size_t)img * BINS;
  unsigned loc[8];
  unsigned s = 0;
  #pragma unroll
  for (int k = 0; k < 8; k++) { loc[k] = h[t * 8 + k]; s += loc[k]; }
  part[t] = s;
  if (t == 0) best = 0;
  __syncthreads();
  /* inclusive suffix scan (Hillis-Steele) */
  unsigned v = s;
  for (int d = 1; d < 512; d <<= 1) {
    unsigned add = (t + d < 512) ? part[t + d] : 0u;
    __syncthreads();
    v += add;
    part[t] = v;
    __syncthreads();
  }
  unsigned sufAfter = (t + 1 < 512) ? part[t + 1] : 0u;
  unsigned cum = sufAfter;
  int cand = -1;
  for (int k = 7; k >= 0; k--) {              /* descending bins within chunk */
    cum += loc[k];
    if (cum >= (unsigned)PRE_K) { cand = t * 8 + k; break; }
  }
  if (cand >= 0) atomicMax(&best, cand);
  __syncthreads();
  if (t == 0) thr[img] = (unsigned)best;
}

/* ---------------- K3: compact candidates above threshold ---------------- */
__global__ __launch_bounds__(256) void k3_compact(const float* __restrict__ obj,
                                                  const unsigned* __restrict__ thr,
                                                  unsigned* __restrict__ cnt,
                                                  unsigned long long* __restrict__ cand) {
  const int img = blockIdx.y;
  const unsigned T = thr[img];
  const float4* src = (const float4*)(obj + (size_t)img * A_ANCH);
  unsigned long long* dst = cand + (size_t)img * CAND;
  const int n4 = A_ANCH / 4;
  const int stride = gridDim.x * blockDim.x;
  for (int i = blockIdx.x * blockDim.x + threadIdx.x; i < n4; i += stride) {
    __builtin_prefetch(src + i + 2 * stride, 0, 0);
    float4 v = src[i];
    float e[4] = {v.x, v.y, v.z, v.w};
    #pragma unroll
    for (int c = 0; c < 4; c++) {
      unsigned k = key_of(e[c]);
      if ((k >> 20) >= T) {
        unsigned p = atomicAdd(&cnt[img], 1u);
        if (p < CAND)
          dst[p] = (((unsigned long long)k) << 32) | (unsigned)(4 * i + c);
      }
    }
  }
}

/* ---------------- K4: async-stage, bitonic sort, decode top-2000 ---------------- */
__global__ __launch_bounds__(1024) void k4_sortdecode(
    const float* __restrict__ deltas, const float* __restrict__ anchors,
    const unsigned long long* __restrict__ cand, const unsigned* __restrict__ cnt,
    float4* __restrict__ boxesS, float* __restrict__ scoresS,
    unsigned* __restrict__ keepW) {
  __shared__ unsigned long long s_arr[CAND];     /* 64 KB */
  const int img = blockIdx.x, tid = threadIdx.x;
  unsigned cc = cnt[img];
  const int c = (cc > (unsigned)CAND) ? CAND : (int)cc;
  const unsigned long long* g = cand + (size_t)img * CAND;

  for (int s = tid; s < CAND; s += 1024)
    if (s >= c) s_arr[s] = 0ull;                 /* pad = smallest key */
#if HAVE_ASYNC_LDS
  for (int s = tid; s < c; s += 1024)
    __builtin_amdgcn_global_load_async_to_lds_b64(
        (AS1 async_v2i*)(g + s),
        (AS3 async_v2i*)(&s_arr[s]), 0, 0);
  __builtin_amdgcn_s_wait_asynccnt(0);
#else
  for (int s = tid; s < c; s += 1024) s_arr[s] = g[s];
#endif
  __syncthreads();

  /* bitonic sort, descending by (key,idx) packed u64 */
  for (int k = 2; k <= CAND; k <<= 1) {
    for (int j = k >> 1; j > 0; j >>= 1) {
      for (int idx = tid; idx < CAND; idx += 1024) {
        int ixj = idx ^ j;
        if (ixj > idx) {
          unsigned long long a = s_arr[idx], b = s_arr[ixj];
          bool desc = ((idx & k) == 0);
          if (desc ? (a < b) : (a > b)) { s_arr[idx] = b; s_arr[ixj] = a; }
        }
      }
      __syncthreads();
    }
  }

  /* decode + clip + validity ballot (branch-free across the 2000 boundary) */
  for (int t = tid; t < K_PAD; t += 1024) {
    const bool inR = t < PRE_K;
    unsigned long long e = inR ? s_arr[t] : 0ull;
    unsigned key = (unsigned)(e >> 32);
    unsigned ai = (unsigned)e;                    /* 0 when !inR: safe gather */
    float logit = key_to_float(key);
    float score = 1.0f / (1.0f + __expf(-logit));
    float4 an = ((const float4*)anchors)[ai];
    const float* dl = deltas + ((size_t)img * A_ANCH + ai) * 4;
    float dx = dl[0], dy = dl[1];
    float dw = fminf(dl[2], BBOX_CLIP), dh = fminf(dl[3], BBOX_CLIP);
    float wa = an.z - an.x, ha = an.w - an.y;
    float cxa = an.x + 0.5f * wa, cya = an.y + 0.5f * ha;
    float pcx = dx * wa + cxa, pcy = dy * ha + cya;
    float pw = __expf(dw) * wa, ph = __expf(dh) * ha;
    float x1 = fminf(fmaxf(pcx - 0.5f * pw, 0.f), IMG_SZ);
    float y1 = fminf(fmaxf(pcy - 0.5f * ph, 0.f), IMG_SZ);
    float x2 = fminf(fmaxf(pcx + 0.5f * pw, 0.f), IMG_SZ);
    float y2 = fminf(fmaxf(pcy + 0.5f * ph, 0.f), IMG_SZ);
    bool valid = inR && ((x2 - x1) >= MIN_SZ) && ((y2 - y1) >= MIN_SZ) && (score > 0.0f);
    unsigned long long bal = __ballot((int)valid);   /* wave32: low 32 bits */
    boxesS[(size_t)img * K_PAD + t] = inR ? make_float4(x1, y1, x2, y2)
                                          : make_float4(0.f, 0.f, 0.f, 0.f);
    scoresS[(size_t)img * K_PAD + t] = inR ? score : 0.f;
    if ((t & 31) == 0) keepW[img * SUPW + (t >> 5)] = (unsigned)bal;
  }
}

/* ---------------- K5: suppression bit-matrix ---------------- */
__global__ __launch_bounds__(64) void k5_supmat(const float4* __restrict__ boxesS,
                                                unsigned* __restrict__ sup) {
  const int i = blockIdx.x;        /* row 0..1999 */
  const int img = blockIdx.y;
  const int cw = threadIdx.x;      /* word 0..63 */
  const float4* B = boxesS + (size_t)img * K_PAD;
  float4 bi = B[i];
  float area_i = (bi.z - bi.x) * (bi.w - bi.y);
  unsigned bits = 0u;
  const int j0 = cw * 32;
  #pragma unroll 4
  for (int b = 0; b < 32; b++) {
    int j = j0 + b;
    float4 bj = B[j];
    float xx1 = fmaxf(bi.x, bj.x), yy1 = fmaxf(bi.y, bj.y);
    float xx2 = fminf(bi.z, bj.z), yy2 = fminf(bi.w, bj.w);
    float inter = fmaxf(xx2 - xx1, 0.f) * fmaxf(yy2 - yy1, 0.f);
    float area_j = (bj.z - bj.x) * (bj.w - bj.y);
    bool s = (j > i) && (j < PRE_K) &&
             (inter > IOU_THR * (area_i + area_j - inter + 1e-12f));
    bits |= (s ? 1u : 0u) << b;
  }
  sup[((size_t)img * K_PAD + i) * SUPW + cw] = bits;
}

/* ---------------- K6: single-wave sequential NMS scan ---------------- */
__global__ __launch_bounds__(32) void k6_scan(const unsigned* __restrict__ sup,
                                              unsigned* __restrict__ keepW) {
  const int img = blockIdx.x;
  const int lane = threadIdx.x;          /* 0..31; owns words lane and lane+32 */
  unsigned r0 = keepW[img * SUPW + lane];
  unsigned r1 = keepW[img * SUPW + lane + 32];
  const unsigned* base = sup + (size_t)img * K_PAD * SUPW;
  const int PF = 8;
  unsigned b0[PF], b1[PF];
  #pragma unroll
  for (int p = 0; p < PF; p++) {
    b0[p] = base[(size_t)p * SUPW + lane];
    b1[p] = base[(size_t)p * SUPW + lane + 32];
  }
  for (int ii = 0; ii < K_PAD; ii += PF) {
    #pragma unroll
    for (int p = 0; p < PF; p++) {
      int i = ii + p;
      unsigned s0 = b0[p], s1 = b1[p];
      int nr = i + PF;
      if (nr < K_PAD) {                  /* pipelined prefetch of row i+PF */
        b0[p] = base[(size_t)nr * SUPW + lane];
        b1[p] = base[(size_t)nr * SUPW + lane + 32];
      }
      if (i < PRE_K) {
        int w = i >> 5, bit = i & 31;
        unsigned wrd = (w < 32) ? (unsigned)__shfl((int)r0, w)
                                : (unsigned)__shfl((int)r1, w - 32);
        if ((wrd >> bit) & 1u) { r0 &= ~s0; r1 &= ~s1; }
      }
    }
  }
  keepW[img * SUPW + lane] = r0;
  keepW[img * SUPW + lane + 32] = r1;
}

/* ---------------- K7: re-pack kept-first, emit [N,1000,5] ----------------
 * Word-level exclusive prefix of the 64 popcounts is computed with
 * V_WMMA_F32_16X16X4_F32: A(16x4)=counts, B[k][n]=(k<=(n&3)) -> one matmul
 * yields all intra-group-of-4 inclusive prefixes + all 16 group sums.     */
__global__ __launch_bounds__(256) void k7_final(const float4* __restrict__ boxesS,
                                                const float* __restrict__ scoresS,
                                                const unsigned* __restrict__ keepW,
                                                float* __restrict__ out) {
  __shared__ unsigned w[SUPW];
  __shared__ float pcnt[SUPW];
  __shared__ float intra[SUPW];    /* intra[m*4+p] = sum pcnt[4m..4m+p]    */
  __shared__ float gex[17];        /* exclusive group prefix; [16] = total */
  __shared__ unsigned pref[SUPW];  /* per-word exclusive prefix            */
  const int img = blockIdx.x, tid = threadIdx.x;
  if (tid < SUPW) {
    w[tid] = keepW[img * SUPW + tid];
    pcnt[tid] = (float)__popc(w[tid]);
  }
  __syncthreads();
#if HAVE_WMMA4
  if (tid < 32) {                  /* wave 0 only: EXEC all-ones for WMMA */
    const int L = tid;
    const int M = L & 15;
    const int K0 = (L >> 4) * 2;   /* A: lanes 0-15 hold K=0,1; 16-31 K=2,3 */
    v2f a; a.x = pcnt[4 * M + K0]; a.y = pcnt[4 * M + K0 + 1];
    const int nn = L & 3;          /* (col n)&3 */
    v2f b; b.x = (K0 <= nn) ? 1.f : 0.f; b.y = ((K0 + 1) <= nn) ? 1.f : 0.f;
    v8f cz = {};
    v8f d = __builtin_amdgcn_wmma_f32_16x16x4_f32(false, a, false, b,
                                                  (short)0, cz, false, false);
    if ((L & 15) < 4) {            /* columns 0..3 carry the 4 prefixes */
      const int mbase = (L >= 16) ? 8 : 0;
      const int p = L & 15;
      #pragma unroll
      for (int r = 0; r < 8; r++) intra[(mbase + r) * 4 + p] = d[r];
    }
  }
#else
  if (tid < 16) {
    float s = 0.f;
    #pragma unroll
    for (int p = 0; p < 4; p++) { s += pcnt[tid * 4 + p]; intra[tid * 4 + p] = s; }
  }
#endif
  __syncthreads();
  if (tid == 0) {
    float s = 0.f;
    #pragma unroll
    for (int m = 0; m < 16; m++) { gex[m] = s; s += intra[m * 4 + 3]; }
    gex[16] = s;
  }
  __syncthreads();
  if (tid < SUPW) {
    float ex = gex[tid >> 2] +
               ((tid & 3) ? intra[(tid >> 2) * 4 + (tid & 3) - 1] : 0.f);
    pref[tid] = (unsigned)(ex + 0.5f);
  }
  __syncthreads();
  const unsigned nk = (unsigned)(gex[16] + 0.5f);
  for (int t = tid; t < PRE_K; t += 256) {
    unsigned word = w[t >> 5];
    int bit = t & 31;
    unsigned pk = pref[t >> 5] + __popc(word & ((1u << bit) - 1u));
    bool kp = (word >> bit) & 1u;
    unsigned dst = kp ? pk : (nk + (unsigned)t - pk);
    if (dst < (unsigned)POST_K) {
      float4 b = boxesS[(size_t)img * K_PAD + t];
      float sc = kp ? scoresS[(size_t)img * K_PAD + t] : 0.f;
      float* o = out + ((size_t)img * POST_K + dst) * 5;
      o[0] = b.x; o[1] = b.y; o[2] = b.z; o[3] = b.w; o[4] = sc;
    }
  }
}

/* ---------------- workspace layout ---------------- */
#define O_HIST   ((size_t)0)                         /* N*BINS*4   = 131072 */
#define O_THR    ((size_t)131072)                    /* N*4               */
#define O_CNT    ((size_t)131328)                    /* N*4               */
#define O_CAND   ((size_t)131584)                    /* N*CAND*8  = 524288 */
#define O_BOXES  ((size_t)655872)                    /* N*K_PAD*16= 262144 */
#define O_SCORES ((size_t)918016)                    /* N*K_PAD*4 =  65536 */
#define O_KEEP   ((size_t)983552)                    /* N*64*4    =   2048 */
#define O_SUP    ((size_t)985600)                    /* N*K_PAD*64*4 = 4194304 */
#define WS_NEED  ((size_t)5179904)

extern "C" void kernel_launch(void* const* d_in, const int* in_sizes, int n_in,
                              void* d_out, int out_size, void* d_ws, size_t ws_size,
                              hipStream_t stream) {
  (void)in_sizes; (void)n_in; (void)out_size;
  if (ws_size < WS_NEED) return;
  const float* obj = (const float*)d_in[0];        /* [8,1e6]    */
  const float* deltas = (const float*)d_in[1];     /* [8,1e6,4]  */
  const float* anchors = (const float*)d_in[2];    /* [1e6,4]    */
  float* out = (float*)d_out;                      /* [8,1000,5] */
  char* ws = (char*)d_ws;
  unsigned* hist = (unsigned*)(ws + O_HIST);
  unsigned* thr  = (unsigned*)(ws + O_THR);
  unsigned* cnt  = (unsigned*)(ws + O_CNT);
  unsigned long long* cand = (unsigned long long*)(ws + O_CAND);
  float4* boxesS = (float4*)(ws + O_BOXES);
  float* scoresS = (float*)(ws + O_SCORES);
  unsigned* keepW = (unsigned*)(ws + O_KEEP);
  unsigned* sup = (unsigned*)(ws + O_SUP);

  k0_init<<<(N_IMG * BINS + 255) / 256, 256, 0, stream>>>(hist, cnt);
  dim3 gs(128, N_IMG);
  k1_hist<<<gs, 256, 0, stream>>>(obj, hist);
  k2_thresh<<<N_IMG, 512, 0, stream>>>(hist, thr);
  k3_compact<<<gs, 256, 0, stream>>>(obj, thr, cnt, cand);
  k4_sortdecode<<<N_IMG, 1024, 0, stream>>>(deltas, anchors, cand, cnt,
                                            boxesS, scoresS, keepW);
  dim3 g5(PRE_K, N_IMG);
  k5_supmat<<<g5, 64, 0, stream>>>(boxesS, sup);
  k6_scan<<<N_IMG, 32, 0, stream>>>(sup, keepW);
  k7_final<<<N_IMG, 256, 0, stream>>>(boxesS, scoresS, keepW, out);
}